// SampleLayer_1314259993031
// MI455X (gfx1250) — compile-verified
//
#include <hip/hip_runtime.h>
#include <stdint.h>

// ---------------------------------------------------------------------------
// retrieval top-k:  logits = x @ kernel^T  ->  per-row top-100 indices (f32)
// softmax is monotonic => rank on raw logits.  labels input unused.
//
// Kernel 1: WMMA fp32 GEMM (V_WMMA_F32_16X16X4_F32) fused with exact
//           streaming top-100 per (16-row block, 16384-item range).
// Kernel 2: 16-way merge of sorted partial lists -> final indices.
// ---------------------------------------------------------------------------

#define B_ROWS    1024
#define EMBED     128
#define NITEMS    262144
#define TOPK      100

#define ROWS_PER_WG 16
#define RANGES      16
#define ITEMS_PER_RANGE (NITEMS / RANGES)     // 16384
#define SUBCHUNK    2048
#define NSUB        (ITEMS_PER_RANGE / SUBCHUNK) // 8
#define CAND_CAP    256
#define LOG_PITCH   2052                      // 2048 + 4 pad -> conflict-free
#define XT_PITCH    132                       // 128 + 4 pad

typedef __attribute__((ext_vector_type(2))) float v2f;
typedef __attribute__((ext_vector_type(8))) float v8f;
typedef unsigned long long u64;
typedef unsigned int       u32;

// monotone float -> uint mapping (bigger float => bigger uint)
__device__ __forceinline__ u32 flip_f32(float f) {
  u32 u = __float_as_uint(f);
  return (u & 0x80000000u) ? ~u : (u | 0x80000000u);
}
// sortable key: value desc, ties -> smaller index wins (matches jax top_k)
__device__ __forceinline__ u64 make_key(float v, u32 idx) {
  return ((u64)flip_f32(v) << 32) | (u64)(0xFFFFFFFFu - idx);
}
__device__ __forceinline__ u32 key_to_idx(u64 k) {
  return 0xFFFFFFFFu - (u32)(k & 0xFFFFFFFFull);
}
// 64-bit shfl_xor built from two 32-bit shuffles (overload-safe)
__device__ __forceinline__ u64 shfl_xor_u64(u64 v, int m) {
  u32 lo = (u32)v, hi = (u32)(v >> 32);
  lo = (u32)__shfl_xor((int)lo, m, 32);
  hi = (u32)__shfl_xor((int)hi, m, 32);
  return ((u64)hi << 32) | (u64)lo;
}

__global__ __launch_bounds__(256, 1)
void gemm_topk_partial(const float* __restrict__ x,
                       const float* __restrict__ kern,
                       u64* __restrict__ partial) {
  __shared__ float xT[ROWS_PER_WG][XT_PITCH];
  __shared__ float logitsS[ROWS_PER_WG][LOG_PITCH];
  __shared__ u64   curList[ROWS_PER_WG][TOPK];
  __shared__ u64   newList[ROWS_PER_WG][TOPK];
  __shared__ u64   cand[ROWS_PER_WG][CAND_CAP];
  __shared__ int   candCnt[ROWS_PER_WG];

  const int tid  = threadIdx.x;
  const int wave = tid >> 5;
  const int lane = tid & 31;
  const int hi   = lane >> 4;      // half-wave select (K pair / M upper-8)
  const int lo   = lane & 15;      // M (A) / N (B,C)

  const int rowBase   = blockIdx.y * ROWS_PER_WG;
  const int rangeBase = blockIdx.x * ITEMS_PER_RANGE;

  // ---- stage x tile (16 x 128) into LDS, coalesced ----
  for (int i = tid; i < ROWS_PER_WG * EMBED; i += 256)
    xT[i >> 7][i & 127] = x[(size_t)(rowBase + (i >> 7)) * EMBED + (i & 127)];
  __syncthreads();

  // ---- hoist all 32 A fragments (K=128 / 4): constant for whole kernel ----
  // A 16x4 fp32 layout: lane(lo)=M, VGPR pair = K {4s+2*hi, 4s+2*hi+1}
  v2f afrag[32];
#pragma unroll
  for (int s = 0; s < 32; ++s)
    afrag[s] = *(const v2f*)&xT[lo][4 * s + 2 * hi];

  const int selRow = tid >> 4;     // 0..15 : each row owned by one half-wave
  const int l16    = tid & 15;

  for (int sub = 0; sub < NSUB; ++sub) {
    const int subBase = rangeBase + sub * SUBCHUNK;
    __syncthreads();               // logits tile free to overwrite

    // ---------------- GEMM: 16 (16x16) tiles per wave -------------------
    for (int t = 0; t < SUBCHUNK / (16 * 8); ++t) {
      const int tileCol = (t * 8 + wave) * 16;
      // B 4x16 fp32: lane(lo)=N(item), VGPR pair = K {4s+2*hi, +1}
      const float* bPtr = kern + (size_t)(subBase + tileCol + lo) * EMBED + 2 * hi;
      v8f acc = {0.f, 0.f, 0.f, 0.f, 0.f, 0.f, 0.f, 0.f};
#pragma unroll
      for (int s = 0; s < 32; ++s) {
        v2f b = *(const v2f*)(bPtr + 4 * s);
        acc = __builtin_amdgcn_wmma_f32_16x16x4_f32(
            /*neg_a=*/false, afrag[s], /*neg_b=*/false, b,
            /*c_mod=*/(short)0, acc, /*reuse_a=*/false, /*reuse_b=*/false);
      }
      // prefetch this lane's item row for the wave's next tile (+128 items)
      __builtin_prefetch(bPtr + 128 * EMBED, 0, 1);
      // C/D: lane(lo)=N, M = r + 8*hi
#pragma unroll
      for (int r = 0; r < 8; ++r)
        logitsS[r + 8 * hi][tileCol + lo] = acc[r];
    }
    __syncthreads();

    // ---------------- exact top-100 selection ---------------------------
    if (sub == 0) {
      // full 100-pass argmax over 2048 logits (16 lanes per row)
      for (int k = 0; k < TOPK; ++k) {
        u64 best = 0;
        for (int c = l16; c < SUBCHUNK; c += 16) {
          u64 key = make_key(logitsS[selRow][c], (u32)(subBase + c));
          if (key > best) best = key;
        }
#pragma unroll
        for (int m = 8; m > 0; m >>= 1) {
          u64 o = shfl_xor_u64(best, m);
          if (o > best) best = o;
        }
        if (l16 == 0) {
          curList[selRow][k] = best;
          int c = (int)(key_to_idx(best)) - subBase;
          logitsS[selRow][c] = -__builtin_inff();  // consume winner
        }
      }
    } else {
      // threshold filter against running 100th-best, then merge survivors
      const u64 thresh = curList[selRow][TOPK - 1];
      if (l16 == 0) candCnt[selRow] = 0;
      for (int c = l16; c < SUBCHUNK; c += 16) {
        u64 key = make_key(logitsS[selRow][c], (u32)(subBase + c));
        if (key > thresh) {
          int slot = atomicAdd(&candCnt[selRow], 1);
          if (slot < CAND_CAP) cand[selRow][slot] = key;
        }
      }
      int cnt = candCnt[selRow];
      if (cnt > CAND_CAP) cnt = CAND_CAP;
      if (cnt > 0) {
        const int tot = TOPK + cnt;
        for (int k = 0; k < TOPK; ++k) {
          u64 best = 0;
          for (int c = l16; c < tot; c += 16) {
            u64 key = (c < TOPK) ? curList[selRow][c] : cand[selRow][c - TOPK];
            if (key > best) best = key;
          }
#pragma unroll
          for (int m = 8; m > 0; m >>= 1) {
            u64 o = shfl_xor_u64(best, m);
            if (o > best) best = o;
          }
          if (l16 == 0) newList[selRow][k] = best;
          // consume winner (keys are unique: index embedded)
          for (int c = l16; c < tot; c += 16) {
            if (c < TOPK) {
              if (curList[selRow][c] == best) curList[selRow][c] = 0;
            } else {
              if (cand[selRow][c - TOPK] == best) cand[selRow][c - TOPK] = 0;
            }
          }
        }
        for (int k = l16; k < TOPK; k += 16)
          curList[selRow][k] = newList[selRow][k];
      }
    }
  }

  __syncthreads();
  const size_t base =
      ((size_t)(rowBase + selRow) * RANGES + (size_t)blockIdx.x) * TOPK;
  for (int k = l16; k < TOPK; k += 16)
    partial[base + k] = curList[selRow][k];
}

__global__ __launch_bounds__(64)
void merge_topk(const u64* __restrict__ partial, float* __restrict__ out) {
  __shared__ u64 keys[RANGES * TOPK];
  const int row = blockIdx.x;
  for (int i = threadIdx.x; i < RANGES * TOPK; i += blockDim.x)
    keys[i] = partial[(size_t)row * RANGES * TOPK + i];
  __syncthreads();
  if (threadIdx.x == 0) {
    int ptr[RANGES];
#pragma unroll
    for (int L = 0; L < RANGES; ++L) ptr[L] = 0;
    for (int k = 0; k < TOPK; ++k) {
      u64 best = 0; int bl = 0;
      for (int L = 0; L < RANGES; ++L) {
        if (ptr[L] < TOPK) {
          u64 kk = keys[L * TOPK + ptr[L]];
          if (kk > best) { best = kk; bl = L; }
        }
      }
      ptr[bl]++;
      out[(size_t)row * TOPK + k] = (float)key_to_idx(best);
    }
  }
}

extern "C" void kernel_launch(void* const* d_in, const int* in_sizes, int n_in,
                              void* d_out, int out_size, void* d_ws, size_t ws_size,
                              hipStream_t stream) {
  (void)in_sizes; (void)n_in; (void)out_size; (void)ws_size;
  const float* x    = (const float*)d_in[0];   // [1024,128] f32
  // d_in[1] = labels (unused on inference branch)
  const float* kern = (const float*)d_in[2];   // [262144,128] f32
  u64* partial = (u64*)d_ws;                   // 1024*16*100*8B = 12.8 MB

  dim3 g1(RANGES, B_ROWS / ROWS_PER_WG);       // (16, 64)
  hipLaunchKernelGGL(gemm_topk_partial, g1, dim3(256), 0, stream,
                     x, kern, partial);
  hipLaunchKernelGGL(merge_topk, dim3(B_ROWS), dim3(64), 0, stream,
                     partial, (float*)d_out);
}